// SwitchTransformerLayer_69982197121266
// MI455X (gfx1250) — compile-verified
//
#include <hip/hip_runtime.h>
#include <math.h>

typedef float v2f __attribute__((ext_vector_type(2)));
typedef float v8f __attribute__((ext_vector_type(8)));
typedef unsigned int u32x4 __attribute__((ext_vector_type(4)));
typedef int i32x8 __attribute__((ext_vector_type(8)));
typedef int i32x4 __attribute__((ext_vector_type(4)));

#define AS1 __attribute__((address_space(1)))
#define AS3 __attribute__((address_space(3)))

#define NTOK 8192
#define DIM  1024
#define FFN  4096
#define OUTD 1024
#define NEXP 8

// X tile layout: element (r,c) at dword offset r*1032 + c + 2*(c>>8)
// == linear idx + 2*(idx>>8): exactly what TDM pad_interval=256dw/pad_amount=2dw
// produces. Effective row stride 1032 (mod 64 = 8) -> <=2-way bank conflict on
// b64 A-fragment reads, b64 alignment preserved (pads are 2 dwords).
#define XS_ELEMS (16 * 1032)
#define HS_STRIDE 132    // 128 + 4 pad: conflict-free b64 A-frag reads

__device__ __forceinline__ int xoff(int r, int c) {
    return r * 1032 + c + 2 * (c >> 8);
}

// staging tier selection
#if __has_builtin(__builtin_amdgcn_tensor_load_to_lds) && __has_builtin(__builtin_amdgcn_s_wait_tensorcnt)
  #define XSTAGE_TDM 1
#elif __has_builtin(__builtin_amdgcn_global_load_async_to_lds_b64) && __has_builtin(__builtin_amdgcn_s_wait_asynccnt)
  #define XSTAGE_ASYNC 1
#endif

// ---------------- init: zero atomic counters ----------------
__global__ void init_kernel(int* ws_ints) {
    if (threadIdx.x < 32) ws_ints[threadIdx.x] = 0;
}

// ---------------- router: logits -> top1 + counts ----------------
__global__ void router_kernel(const float* __restrict__ x,
                              const float* __restrict__ Wg,
                              const float* __restrict__ bg,
                              int* __restrict__ top1,
                              int* __restrict__ counts) {
    int n = blockIdx.x * blockDim.x + threadIdx.x;
    if (n >= NTOK) return;
    float acc[NEXP];
#pragma unroll
    for (int e = 0; e < NEXP; ++e) acc[e] = bg[e];
    const float* xr = x + (size_t)n * DIM;
    for (int d = 0; d < DIM; d += 4) {
        float4 xv = *(const float4*)(xr + d);
        const float* w0 = Wg + (size_t)d * NEXP;
#pragma unroll
        for (int e = 0; e < NEXP; ++e) {
            acc[e] += xv.x * w0[e] + xv.y * w0[NEXP + e]
                    + xv.z * w0[2 * NEXP + e] + xv.w * w0[3 * NEXP + e];
        }
    }
    int best = 0;
    float bv = acc[0];
#pragma unroll
    for (int e = 1; e < NEXP; ++e) {
        if (acc[e] > bv) { bv = acc[e]; best = e; }   // first-max tie rule == argmax
    }
    top1[n] = best;
    atomicAdd(&counts[best], 1);
}

// ---------------- scan + aux loss ----------------
__global__ void scan_aux_kernel(const int* __restrict__ counts,
                                int* __restrict__ offsets,
                                float* __restrict__ out_tail) {
    if (threadIdx.x != 0) return;
    int sum = 0;
    int c[NEXP];
#pragma unroll
    for (int e = 0; e < NEXP; ++e) { c[e] = counts[e]; offsets[e] = sum; sum += c[e]; }
    float total = (float)sum;
    const float u = 1.0f / (float)NEXP + 1e-10f;
    float aux = 0.0f;
#pragma unroll
    for (int e = 0; e < NEXP; ++e) {
        float p = ((float)c[e] + 1e-10f) / total;
        aux += u * (logf(u) - logf(p));
    }
    aux /= (float)NEXP;
    out_tail[0] = 0.01f * aux;                 // AUX_LOSS_WEIGHT * aux
#pragma unroll
    for (int e = 0; e < NEXP; ++e) out_tail[1 + e] = (float)c[e];
}

// ---------------- scatter tokens into per-expert lists ----------------
__global__ void scatter_kernel(const int* __restrict__ top1,
                               const int* __restrict__ offsets,
                               int* __restrict__ fill,
                               int* __restrict__ token_ids) {
    int n = blockIdx.x * blockDim.x + threadIdx.x;
    if (n >= NTOK) return;
    int e = top1[n];
    int pos = atomicAdd(&fill[e], 1);
    token_ids[offsets[e] + pos] = n;
}

// ---------------- grouped expert FFN: WMMA f32 + TDM gather ----------------
__global__ __launch_bounds__(256)
void ffn_kernel(const float* __restrict__ x,
                const float* __restrict__ W1,
                const float* __restrict__ b1,
                const float* __restrict__ W2,
                const float* __restrict__ b2,
                const int* __restrict__ token_ids,
                const int* __restrict__ offsets,
                const int* __restrict__ counts,
                float* __restrict__ out) {
    __shared__ float Xs[XS_ELEMS];
    __shared__ float Hs[16 * HS_STRIDE];
    __shared__ int   tokS[16];

    const int e    = blockIdx.x >> 9;     // 512 tiles per expert
    const int tile = blockIdx.x & 511;
    const int cnt  = counts[e];
    const int gbase = tile * 16;
    if (gbase >= cnt) return;             // uniform exit
    const int off = offsets[e];
    const int mvalid = min(16, cnt - gbase);

    const int tid  = threadIdx.x;
    const int wave = tid >> 5;            // 0..7
    const int lane = tid & 31;
    const int half = lane >> 4;           // 0 or 1
    const int l16  = lane & 15;

    if (tid < 16) {
        int g = gbase + tid;
        int gi = (g < cnt) ? g : (cnt - 1);
        tokS[tid] = token_ids[off + gi];
    }
    __syncthreads();

    // ---- stage X tile [16 x 1024] into LDS (padded layout) ----
#if defined(XSTAGE_TDM)
    if (wave == 0) {
        // Tensor DMA gather: 16 rows of 1024 f32, row indices = token ids,
        // LDS padding 2 dwords per 256 dwords -> row stride 1032.
        unsigned ldsa = (unsigned)(unsigned long long)(AS3 void*)&Xs[0];
        unsigned long long ga = (unsigned long long)(const void*)x;
        u32x4 g0;
        g0[0] = 0x80000001u;                          // count=1, gather_mode=1, 16-bit idx
        g0[1] = ldsa;                                 // lds_addr
        g0[2] = (unsigned)(ga & 0xFFFFFFFFu);         // global_addr[31:0]
        g0[3] = (unsigned)((ga >> 32) & 0x01FFFFFFu)  // global_addr[56:32]
              | 0x80000000u;                          // type=2 ("image")
        i32x8 g1;
        g1[0] = 0x03D20000;   // data_size=4B, pad_enable, interval=256dw, amount=2dw
        g1[1] = 0x04000000;   // tensor_dim0 = 1024 (bits[79:48] low half)
        g1[2] = 0x20000000;   // tensor_dim1 = 8192 (bits[111:80] low half)
        g1[3] = 0x04000000;   // tile_dim0 = 1024 (bits[127:112])
        g1[4] = 16;           // tile_dim1 = 16 valid gather indices
        g1[5] = 1024;         // tensor_dim0_stride (data_size units)
        g1[6] = 0;
        g1[7] = 0;
        i32x4 g2, g3;
#pragma unroll
        for (int i2 = 0; i2 < 4; ++i2) {
            g2[i2] = (tokS[2 * i2] & 0xFFFF) | (tokS[2 * i2 + 1] << 16);
            g3[i2] = (tokS[8 + 2 * i2] & 0xFFFF) | (tokS[9 + 2 * i2] << 16);
        }
        i32x8 g4;             // VADDR4 slot: unused per ISA, zeroed
#pragma unroll
        for (int i4 = 0; i4 < 8; ++i4) g4[i4] = 0;
        __builtin_amdgcn_tensor_load_to_lds(g0, g1, g2, g3, g4, 0);
        __builtin_amdgcn_s_wait_tensorcnt(0);
    }
    __syncthreads();
#elif defined(XSTAGE_ASYNC)
    for (int idx = tid; idx < 16 * (DIM / 2); idx += 256) {
        int r = idx >> 9, c2 = idx & 511, c = c2 * 2;
        const float* gp = x + (size_t)tokS[r] * DIM + c;
        float* lp = &Xs[xoff(r, c)];
        __builtin_amdgcn_global_load_async_to_lds_b64(
            (AS1 const void*)gp, (AS3 void*)lp, 0, 0);
    }
    __builtin_amdgcn_s_wait_asynccnt(0);
    __syncthreads();
#else
    for (int idx = tid; idx < 16 * (DIM / 2); idx += 256) {
        int r = idx >> 9, c2 = idx & 511, c = c2 * 2;
        float2 xv = ((const float2*)(x + (size_t)tokS[r] * DIM))[c2];
        *(float2*)&Xs[xoff(r, c)] = xv;
    }
    __syncthreads();
#endif

    const float* W1e = W1 + (size_t)e * DIM * FFN;
    const float* b1e = b1 + (size_t)e * FFN;
    const float* W2e = W2 + (size_t)e * FFN * OUTD;
    const float* b2e = b2 + (size_t)e * OUTD;

    v8f yacc[8];
#pragma unroll
    for (int t = 0; t < 8; ++t) yacc[t] = (v8f)(0.0f);

    for (int fc = 0; fc < FFN; fc += 128) {
        // ---- GEMM1: this wave computes H tile cols [fc+16*wave, +16), K = 1024
        const int fcol = fc + wave * 16 + l16;
        v8f hc = (v8f)(0.0f);
        const int xrow = l16 * 1032;
        for (int k = 0; k < DIM; k += 4) {
            int kk = k + 2 * half;
            v2f a = *(const v2f*)&Xs[xrow + kk + 2 * (kk >> 8)];
            const float* w1p = W1e + (size_t)kk * FFN + fcol;
            v2f b;
            b.x = w1p[0];
            b.y = w1p[FFN];
            hc = __builtin_amdgcn_wmma_f32_16x16x4_f32(false, a, false, b,
                                                       (short)0, hc, false, false);
        }
        // bias + exact GELU, store H tile to LDS
        {
            float bv = b1e[fcol];
#pragma unroll
            for (int j = 0; j < 8; ++j) {
                float v = hc[j] + bv;
                v = 0.5f * v * (1.0f + erff(v * 0.70710678118654752440f));
                int m = j + 8 * half;
                Hs[m * HS_STRIDE + wave * 16 + l16] = v;
            }
        }
        __syncthreads();

        // ---- GEMM2: y[16, 128 cols of this wave] += H[16,128] @ W2[fc:fc+128, cols]
        for (int k = 0; k < 128; k += 4) {
            v2f a = *(const v2f*)&Hs[l16 * HS_STRIDE + k + 2 * half];
            const float* w2p = W2e + (size_t)(fc + k + 2 * half) * OUTD
                             + wave * 128 + l16;
#pragma unroll
            for (int t = 0; t < 8; ++t) {
                v2f b;
                b.x = w2p[t * 16];
                b.y = w2p[OUTD + t * 16];
                yacc[t] = __builtin_amdgcn_wmma_f32_16x16x4_f32(false, a, false, b,
                                                                (short)0, yacc[t],
                                                                false, false);
            }
        }
        __syncthreads();   // Hs reused next chunk
    }

    // ---- epilogue: +b2, scatter rows back to token positions
#pragma unroll
    for (int t = 0; t < 8; ++t) {
        int col = wave * 128 + t * 16 + l16;
        float bv = b2e[col];
#pragma unroll
        for (int j = 0; j < 8; ++j) {
            int m = j + 8 * half;
            if (m < mvalid) {
                out[(size_t)tokS[m] * OUTD + col] = yacc[t][j] + bv;
            }
        }
    }
}

// ---------------- launcher ----------------
extern "C" void kernel_launch(void* const* d_in, const int* in_sizes, int n_in,
                              void* d_out, int out_size, void* d_ws, size_t ws_size,
                              hipStream_t stream) {
    const float* x  = (const float*)d_in[0];
    const float* Wg = (const float*)d_in[1];
    const float* bg = (const float*)d_in[2];
    const float* W1 = (const float*)d_in[3];
    const float* b1 = (const float*)d_in[4];
    const float* W2 = (const float*)d_in[5];
    const float* b2 = (const float*)d_in[6];
    float* out = (float*)d_out;

    int* ws_ints   = (int*)d_ws;
    int* counts    = ws_ints;          // [8]
    int* fill      = ws_ints + 8;      // [8]
    int* offsets   = ws_ints + 16;     // [8]
    int* top1      = ws_ints + 64;     // [N]
    int* token_ids = top1 + NTOK;      // [N]

    init_kernel<<<1, 32, 0, stream>>>(ws_ints);
    router_kernel<<<NTOK / 256, 256, 0, stream>>>(x, Wg, bg, top1, counts);
    scan_aux_kernel<<<1, 32, 0, stream>>>(counts, offsets,
                                          out + (size_t)NTOK * OUTD);
    scatter_kernel<<<NTOK / 256, 256, 0, stream>>>(top1, offsets, fill, token_ids);
    ffn_kernel<<<NEXP * 512, 256, 0, stream>>>(x, W1, b1, W2, b2,
                                               token_ids, offsets, counts, out);
}